// Cross_attention_38972533243918
// MI455X (gfx1250) — compile-verified
//
#include <hip/hip_runtime.h>

typedef __attribute__((ext_vector_type(2))) float v2f;
typedef __attribute__((ext_vector_type(8))) float v8f;

#define NH 4
#define HD 16
#define SDIM 32
#define HDIM 24
#define WDIM 24
#define HW 576
#define CDIM 64
#define SSTRIDE (HW * CDIM)   /* 36864 floats between s-slices */
#define SCALE 0.25f

// ---------------------------------------------------------------------------
// Spectral attention: one wave per (b, h, w, n) unit. 32x32 attention, hd=16.
// ---------------------------------------------------------------------------
__global__ __launch_bounds__(32) void spectral_kernel(const float* __restrict__ q,
                                                      const float* __restrict__ kv,
                                                      float* __restrict__ out) {
    __shared__ float sc[32 * 34];            // 32x32 scores, padded row stride 34
    const int unit = blockIdx.x;
    const int n = unit & 3;
    const int p = (unit >> 2) % HW;          // h*W + w
    const int b = unit / (4 * HW);
    const int lane = threadIdx.x;
    const int half = lane >> 4;
    const int lrow = lane & 15;

    const size_t base = (size_t)b * SDIM * (size_t)SSTRIDE + (size_t)p * CDIM + (size_t)n * HD;
    const float* qb = q + base;              // Q[s][d] = qb[s*SSTRIDE + d]
    const float* kb = kv + base;             // K=V[s][d] = kb[s*SSTRIDE + d]
    float* ob = out + base;

    // ---- S = (Q K^T) * scale : 32x32 via 2x2 tiles of 16x16, K=16 (4 steps)
    for (int mt = 0; mt < 2; ++mt) {
        for (int nt = 0; nt < 2; ++nt) {
            v8f acc = {};
#pragma unroll
            for (int k0 = 0; k0 < 16; k0 += 4) {
                const int d = k0 + 2 * half;
                v2f a  = *(const v2f*)(qb + (size_t)(16 * mt + lrow) * SSTRIDE + d);
                v2f bv = *(const v2f*)(kb + (size_t)(16 * nt + lrow) * SSTRIDE + d);
                acc = __builtin_amdgcn_wmma_f32_16x16x4_f32(false, a, false, bv,
                                                            (short)0, acc, false, false);
            }
#pragma unroll
            for (int v = 0; v < 8; ++v)
                sc[(16 * mt + v + 8 * half) * 34 + 16 * nt + lrow] = acc[v] * SCALE;
        }
    }
    __syncthreads();

    // ---- softmax: lane r owns row r (32 rows, 32 lanes)
    {
        float* row = sc + lane * 34;
        float m = row[0];
        for (int j = 1; j < 32; ++j) m = fmaxf(m, row[j]);
        float s = 0.f;
        for (int j = 0; j < 32; ++j) { float e = __expf(row[j] - m); row[j] = e; s += e; }
        const float inv = 1.f / s;
        for (int j = 0; j < 32; ++j) row[j] *= inv;
    }
    __syncthreads();

    // ---- O = P V : 32x16, K=32 (8 steps)
    for (int mt = 0; mt < 2; ++mt) {
        v8f acc = {};
#pragma unroll
        for (int k0 = 0; k0 < 32; k0 += 4) {
            const int kk = k0 + 2 * half;
            v2f a = *(const v2f*)(sc + (16 * mt + lrow) * 34 + kk);
            v2f bv;
            bv.x = kb[(size_t)kk * SSTRIDE + lrow];
            bv.y = kb[(size_t)(kk + 1) * SSTRIDE + lrow];
            acc = __builtin_amdgcn_wmma_f32_16x16x4_f32(false, a, false, bv,
                                                        (short)0, acc, false, false);
        }
#pragma unroll
        for (int v = 0; v < 8; ++v)
            ob[(size_t)(16 * mt + v + 8 * half) * SSTRIDE + lrow] = acc[v];
    }
}

// ---------------------------------------------------------------------------
// Spatial attention: one wave per (b, s, n, qtile) — 16 query rows vs 576 keys.
// Accumulates (+=) into output already written by spectral_kernel.
// ---------------------------------------------------------------------------
__global__ __launch_bounds__(32) void spatial_kernel(const float* __restrict__ q,
                                                     const float* __restrict__ kv,
                                                     float* __restrict__ out) {
    __shared__ float sc[16 * 584];           // 16x576 scores, padded row stride 584
    const int blk = blockIdx.x;
    const int qt = blk % 36;                 // query tile index (16 rows each)
    const int unit = blk / 36;
    const int n = unit & 3;
    const int s = (unit >> 2) % SDIM;
    const int b = unit / (4 * SDIM);
    const int lane = threadIdx.x;
    const int half = lane >> 4;
    const int lrow = lane & 15;

    const size_t base = ((size_t)(b * SDIM + s)) * (size_t)SSTRIDE + (size_t)n * HD;
    const float* qb = q + base + (size_t)qt * 16 * CDIM;  // Q[r][d] = qb[r*C + d]
    const float* kb = kv + base;                           // K=V[p][d] = kb[p*C + d]
    float* ob = out + base + (size_t)qt * 16 * CDIM;

    // ---- scores[16][576] = (Q K^T) * scale
    v2f aq[4];
#pragma unroll
    for (int k0 = 0; k0 < 4; ++k0)
        aq[k0] = *(const v2f*)(qb + (size_t)lrow * CDIM + 4 * k0 + 2 * half);

    for (int nt = 0; nt < 36; ++nt) {
        v8f acc = {};
#pragma unroll
        for (int k0 = 0; k0 < 4; ++k0) {
            v2f bv = *(const v2f*)(kb + (size_t)(16 * nt + lrow) * CDIM + 4 * k0 + 2 * half);
            acc = __builtin_amdgcn_wmma_f32_16x16x4_f32(false, aq[k0], false, bv,
                                                        (short)0, acc, false, false);
        }
#pragma unroll
        for (int v = 0; v < 8; ++v)
            sc[(v + 8 * half) * 584 + 16 * nt + lrow] = acc[v] * SCALE;
    }
    __syncthreads();

    // ---- softmax: 16 rows, 2 workers per row (halves split the 576 columns)
    {
        float* row = sc + lrow * 584;
        const int j0 = half * 288;
        float m = -1e30f;
        for (int j = j0; j < j0 + 288; ++j) m = fmaxf(m, row[j]);
        m = fmaxf(m, __shfl_xor(m, 16));
        float sum = 0.f;
        for (int j = j0; j < j0 + 288; ++j) { float e = __expf(row[j] - m); row[j] = e; sum += e; }
        sum += __shfl_xor(sum, 16);
        const float inv = 1.f / sum;
        for (int j = j0; j < j0 + 288; ++j) row[j] *= inv;
    }
    __syncthreads();

    // ---- O[16][16] = P[16][576] @ V[576][16], K=576 (144 steps)
    v8f acc = {};
    for (int k0 = 0; k0 < 576; k0 += 4) {
        const int kk = k0 + 2 * half;
        v2f a = *(const v2f*)(sc + lrow * 584 + kk);
        v2f bv;
        bv.x = kb[(size_t)kk * CDIM + lrow];
        bv.y = kb[(size_t)(kk + 1) * CDIM + lrow];
        acc = __builtin_amdgcn_wmma_f32_16x16x4_f32(false, a, false, bv,
                                                    (short)0, acc, false, false);
    }
#pragma unroll
    for (int v = 0; v < 8; ++v)
        ob[(size_t)(v + 8 * half) * CDIM + lrow] += acc[v];
}

extern "C" void kernel_launch(void* const* d_in, const int* in_sizes, int n_in,
                              void* d_out, int out_size, void* d_ws, size_t ws_size,
                              hipStream_t stream) {
    (void)in_sizes; (void)n_in; (void)out_size; (void)d_ws; (void)ws_size;
    const float* q  = (const float*)d_in[0];
    const float* kv = (const float*)d_in[1];
    float* out = (float*)d_out;

    // Spectral first: overwrites every output element exactly once.
    spectral_kernel<<<2 * HW * NH, 32, 0, stream>>>(q, kv, out);
    // Spatial second (stream-ordered): accumulates += into output.
    spatial_kernel<<<2 * SDIM * NH * 36, 32, 0, stream>>>(q, kv, out);
}